// CTCLoss_20856361189511
// MI455X (gfx1250) — compile-verified
//
#include <hip/hip_runtime.h>
#include <hip/hip_bf16.h>
#include <stdint.h>

// Problem constants (from reference): T=160, N=64, C=6625, S=25
#define CTC_T 160
#define CTC_N 64
#define CTC_C 6625
#define CTC_S 25
#define CTC_L (2 * CTC_S + 1) /* 51 */
#define CTC_NEG (-1.0e5f)

// ---- gfx1250 feature probes (device pass only; host pass takes fallbacks) ----
#if defined(__gfx1250__)
#  if __has_builtin(__builtin_amdgcn_global_load_async_to_lds_b32)
#    define CTC_HAS_ASYNC_LDS 1
#  endif
#  if __has_builtin(__builtin_amdgcn_wmma_f32_16x16x4_f32)
#    define CTC_HAS_WMMA_F32 1
#  endif
#endif

typedef float v8f __attribute__((ext_vector_type(8)));
typedef float v2f __attribute__((ext_vector_type(2)));

#if defined(__gfx1250__)
typedef __attribute__((address_space(1))) int* gaddr_i32_t;  // global
typedef __attribute__((address_space(3))) int* laddr_i32_t;  // LDS
#endif

__device__ __forceinline__ float lae(float a, float b) {
  // logaddexp matching jnp for finite inputs (NEG=-1e5 keeps everything finite)
  float m = fmaxf(a, b);
  float d = fminf(a, b) - m;
  return m + log1pf(expf(d));
}

// One workgroup (64 threads = 2 waves) per batch element. Lane l owns lattice
// position l (l < 51). alpha lives in LDS with 2 NEG guard cells in front so
// the l-1 / l-2 reads need no branches. Emissions for step t are gathered
// asynchronously into LDS (ASYNCcnt path) when the builtin is available.
__global__ void ctc_alpha_kernel(const float* __restrict__ x,
                                 const int* __restrict__ labels,
                                 const int* __restrict__ seqlen,
                                 float* __restrict__ ws) {
  const int n = blockIdx.x;
  const int l = threadIdx.x;

  __shared__ float sA[CTC_L + 2];  // [0..1] = NEG guards, alpha[l] at [2+l]
  __shared__ float sE[CTC_L];      // async-gathered emissions

  const float* xn = x + (size_t)n * CTC_C;          // x[0, n, :]
  const size_t stepT = (size_t)CTC_N * CTC_C;       // stride between t slices

  // ext[l]: even -> blank(0); odd -> labels[n][(l-1)/2].
  // can_skip[l]: l odd, l>=3, ext[l] != ext[l-2] (blank rows never skip).
  int  extl = 0;
  bool skip = false;
  if (l < CTC_L && (l & 1)) {
    extl = labels[n * CTC_S + (l >> 1)];
    if (l >= 3) skip = (extl != labels[n * CTC_S + ((l - 2) >> 1)]);
  }

  if (l < 2) sA[l] = CTC_NEG;
  if (l < CTC_L) sA[2 + l] = (l < 2) ? xn[extl] : CTC_NEG;  // t = 0 init
  __syncthreads();

  for (int t = 1; t < CTC_T; ++t) {
    const float* src = xn + (size_t)t * stepT + extl;
    float e = 0.0f;
#if CTC_HAS_ASYNC_LDS
    if (l < CTC_L) {
      __builtin_amdgcn_global_load_async_to_lds_b32(
          (gaddr_i32_t)(uintptr_t)src,
          (laddr_i32_t)&sE[l],
          /*offset=*/0, /*cpol=*/0);
    }
    asm volatile("s_wait_asynccnt 0" ::: "memory");
    __syncthreads();
    if (l < CTC_L) e = sE[l];
#else
    if (l < CTC_L) e = *src;
    (void)sE;
#endif
    float r = 0.0f;
    if (l < CTC_L) {
      float a  = sA[2 + l];
      float s1 = sA[1 + l];
      float s2 = sA[l];
      r = lae(a, s1);
      if (skip) r = lae(r, s2);
      r += e;
    }
    __syncthreads();             // all reads of sA done
    if (l < CTC_L) sA[2 + l] = r;
    __syncthreads();             // writes visible before next step's reads
  }

  if (l == 0) {
    int   len  = seqlen[n];
    int   last = 2 * len;                       // final blank position
    float ll   = lae(sA[2 + last], sA[1 + last]);
    ws[n] = -ll / (float)len;                   // per-example loss / length
  }
}

// Batch mean of 64 per-example losses via V_WMMA_F32_16X16X4_F32:
// A = ones(16x4), B carries the 64 losses (one per lane in each of 2 B VGPRs).
// Every row of D equals the B column sums, so summing D-VGPR0 across all 32
// lanes double-counts the total exactly; multiply by 0.5. Fixed-order shuffle
// reduction keeps the result deterministic across replays.
__global__ void ctc_reduce_kernel(const float* __restrict__ ws,
                                  float* __restrict__ out) {
  const int lane = threadIdx.x;  // 32 threads, one wave, EXEC all ones
#if CTC_HAS_WMMA_F32
  v2f a; a.x = 1.0f; a.y = 1.0f;
  v2f b; b.x = ws[lane]; b.y = ws[lane + 32];
  v8f c = {};
  c = __builtin_amdgcn_wmma_f32_16x16x4_f32(
      /*neg_a=*/false, a, /*neg_b=*/false, b,
      /*c_mod=*/(short)0, c, /*reuse_a=*/false, /*reuse_b=*/false);
  float v = c[0];
  for (int off = 16; off > 0; off >>= 1) v += __shfl_xor(v, off, 32);
  if (lane == 0) out[0] = (v * 0.5f) / (float)CTC_N;
#else
  if (lane == 0) {
    float s = 0.0f;
    for (int i = 0; i < CTC_N; ++i) s += ws[i];
    out[0] = s / (float)CTC_N;
  }
#endif
}

extern "C" void kernel_launch(void* const* d_in, const int* in_sizes, int n_in,
                              void* d_out, int out_size, void* d_ws, size_t ws_size,
                              hipStream_t stream) {
  const float* x      = (const float*)d_in[0];  // [T, N, C] log-softmax, f32
  const int*   labels = (const int*)d_in[1];    // [N, S] int32
  const int*   slen   = (const int*)d_in[2];    // [N] int32
  float*       out    = (float*)d_out;          // scalar f32
  float*       ws     = (float*)d_ws;           // >= 64 floats of scratch

  ctc_alpha_kernel<<<CTC_N, 64, 0, stream>>>(x, labels, slen, ws);
  ctc_reduce_kernel<<<1, 32, 0, stream>>>(ws, out);
}